// Pooler_34514357190797
// MI455X (gfx1250) — compile-verified
//
#include <hip/hip_runtime.h>

// Problem constants (baked into the reference module)
#define B_       4
#define C_       256
#define H_       256
#define W_       512
#define MAXLOAD_ 100
#define FS_      14              // feature_size
#define HB_      (B_ * H_)       // 1024 (stacked-batch height of the (C, B*H, W) view)
#define IMH_     1024            // image_height
#define INV_DSCALE_ 0.25f        // 1 / shrink_scale
#define PLANE_   (H_ * W_)       // 131072 elements per (b,c) plane
#define NROI_    (B_ * MAXLOAD_) // 400
#define OUT_MAIN_ ((size_t)NROI_ * C_ * FS_ * FS_)  // 20,070,400 floats

typedef __attribute__((ext_vector_type(2))) float v2f;
typedef __attribute__((ext_vector_type(8))) float v8f;

// One block = (one ROI) x (one group of 4 channel-tiles). 8 waves / block.
// Each wave computes 16x16 output tiles: M = 14(+2 pad) py rows, N = 16 channels,
// accumulating V_WMMA_F32_16X16X4_F32 over 7 chunks of the 28 sampled y-rows.
// All global accesses use a uniform SGPR base + 32-bit per-lane element offset
// (whole tensor spans fit in u32), so loads lower to the saddr+voffset form
// instead of per-lane 64-bit address arithmetic.
__global__ __launch_bounds__(256) void roi_align_wmma(
    const float* __restrict__ features,   // (B, C, H, W) fp32
    const float* __restrict__ rois,       // (B, MAXLOAD, 4) fp32
    float* __restrict__ out)              // (400, 256, 14, 14) fp32
{
  __shared__ float    s_wx[16];
  __shared__ float    s_wy[16];
  __shared__ int      s_x0[16];
  __shared__ unsigned s_row[32];          // element offsets of the 28 sampled rows

  const int n = blockIdx.x;               // ROI index 0..399
  const int b = n / MAXLOAD_;

  // ---- Per-ROI separable grid setup (14 x-columns, 28 y-rows) ----
  if (threadIdx.x < FS_) {
    const int j = (int)threadIdx.x;
    const float4 r = ((const float4*)rois)[n];       // x0, y0, x1, y1
    const float ft = (float)j * (1.0f / (float)(FS_ - 1));

    // x: px = gridx * W/(W-1) - 0.5, clamped  (algebraic reduction of the
    //    normalize->denormalize chain in the reference)
    float gridx = fmaf(ft, r.z - r.x, r.x) * INV_DSCALE_;
    float pxf = gridx * ((float)W_ / (float)(W_ - 1)) - 0.5f;
    pxf = fminf(fmaxf(pxf, 0.0f), (float)(W_ - 1));
    float x0f = floorf(pxf);
    s_wx[j] = pxf - x0f;
    s_x0[j] = (int)x0f;

    // y: includes the batch offset b*image_height; clamp against stacked height
    float biasy = r.y + (float)b * (float)IMH_;
    float gridy = fmaf(ft, r.w - r.y, biasy) * INV_DSCALE_;
    float pyf = gridy * ((float)HB_ / (float)(HB_ - 1)) - 0.5f;
    pyf = fminf(fmaxf(pyf, 0.0f), (float)(HB_ - 1));
    float y0f = floorf(pyf);
    s_wy[j] = pyf - y0f;
    int y0 = (int)y0f;
    int y1 = min(y0 + 1, HB_ - 1);        // may legitimately cross into next batch image
    // flat element offset of row y in the (C, B*H, W) view, minus the c*PLANE_
    // term: (y>>8)*C*H*W + (y&255)*W   (max ~100.8M elements, fits u32)
    s_row[2 * j]     = (unsigned)((y0 >> 8) * (C_ * PLANE_) + (y0 & (H_ - 1)) * W_);
    s_row[2 * j + 1] = (unsigned)((y1 >> 8) * (C_ * PLANE_) + (y1 & (H_ - 1)) * W_);
  }
  __syncthreads();

  const int wave  = (int)threadIdx.x >> 5;
  const int lane  = (int)threadIdx.x & 31;
  const int m     = lane & 15;            // A-row / D-column lane id
  const int khalf = lane >> 4;            // lane-half selects K (and D rows +8)

  // y-weights for this lane's A-matrix row (only consumed when this lane is
  // the active one in a K-chunk; m=14,15 read padded LDS, never used)
  const float aw0 = 1.0f - s_wy[m];
  const float aw1 = s_wy[m];

  // 4 channel-tiles x 14 px = 56 tiles per block, 7 per wave (uniform -> EXEC all-1s)
  for (int t = wave; t < 56; t += 8) {
    const int ctl = t & 3;
    const int px  = t >> 2;
    const int c   = ((int)blockIdx.y * 4 + ctl) * 16 + m;   // this lane's channel

    const int      xa   = s_x0[px];
    const unsigned dx   = (unsigned)(min(xa + 1, W_ - 1) - xa);   // 0 or 1
    const float    wxv  = s_wx[px];
    // per-lane u32 element offset of (channel plane, column xa); rows added per chunk
    const unsigned offx = (unsigned)c * (unsigned)PLANE_ + (unsigned)xa;

    v8f acc = {};
    #pragma unroll
    for (int q = 0; q < 7; ++q) {
      const int kk = 4 * q + 2 * khalf;           // this lane's pair of y-rows
      const unsigned o0 = offx + s_row[kk];
      const unsigned o1 = offx + s_row[kk + 1];
      const float f00 = features[o0];
      const float f01 = features[o0 + dx];
      const float f10 = features[o1];
      const float f11 = features[o1 + dx];

      // B operand: x-direction lerp of the two rows, 16 channels across lanes
      v2f bv;
      bv.x = fmaf(wxv, f01 - f00, f00);
      bv.y = fmaf(wxv, f11 - f10, f10);

      // A operand: sparse y-weight matrix, row m active iff k-chunk covers 2m
      const bool act = (m == 2 * q + khalf);
      v2f av;
      av.x = act ? aw0 : 0.0f;
      av.y = act ? aw1 : 0.0f;

      // D(16x16 f32) += A(16x4 f32) x B(4x16 f32)
      acc = __builtin_amdgcn_wmma_f32_16x16x4_f32(
          /*neg_a=*/false, av, /*neg_b=*/false, bv,
          /*c_mod=*/(short)0, acc, /*reuse_a=*/false, /*reuse_b=*/false);
    }

    // D layout: lane holds column N=m (channel c), rows M = v + 8*khalf
    const unsigned outBase =
        ((unsigned)n * (unsigned)C_ + (unsigned)c) * (unsigned)(FS_ * FS_) + (unsigned)px;
    #pragma unroll
    for (int v = 0; v < 8; ++v) {
      const int py = 8 * khalf + v;
      if (py < FS_) out[outBase + (unsigned)(py * FS_)] = acc[v];
    }
  }
}

__global__ void write_val_bind(float* __restrict__ out) {
  const int i = (int)(blockIdx.x * blockDim.x + threadIdx.x);
  if (i < NROI_) out[OUT_MAIN_ + i] = (float)(i / MAXLOAD_);
}

extern "C" void kernel_launch(void* const* d_in, const int* in_sizes, int n_in,
                              void* d_out, int out_size, void* d_ws, size_t ws_size,
                              hipStream_t stream) {
  const float* features = (const float*)d_in[0];
  const float* rois     = (const float*)d_in[1];
  float* out            = (float*)d_out;

  dim3 grid(NROI_, 4, 1);                 // 1600 blocks x 8 waves = 12800 waves
  roi_align_wmma<<<grid, 256, 0, stream>>>(features, rois, out);
  write_val_bind<<<(NROI_ + 255) / 256, 256, 0, stream>>>(out);
}